// EvolvingGNN_68719477392
// MI455X (gfx1250) — compile-verified
//
#include <hip/hip_runtime.h>
#include <hip/hip_bf16.h>
#include <math.h>

// ---------------- problem constants (from reference) ----------------
#define NN     100000          // nodes
#define NE     1600000         // edges
#define TT     5               // sequence length
#define DIN    32
#define DH     32
#define EF     16
#define FLAT   1024            // DIN*DH
#define KIN    80              // 2*DH+EF  (edge MLP input dim)

typedef float v2f __attribute__((ext_vector_type(2)));
typedef float v8f __attribute__((ext_vector_type(8)));

static __device__ __forceinline__ v8f wmma4(v2f a, v2f b, v8f c) {
  // V_WMMA_F32_16X16X4_F32 : D(16x16,f32) = A(16x4,f32) * B(4x16,f32) + C
  return __builtin_amdgcn_wmma_f32_16x16x4_f32(
      /*neg_a=*/false, a, /*neg_b=*/false, b,
      /*c_mod=*/(short)0, c, /*reuse_a=*/false, /*reuse_b=*/false);
}

static __device__ __forceinline__ float sigmoidf_(float x) {
  return 1.0f / (1.0f + __expf(-x));
}

// ----- async global->LDS copy (gfx1250 GLOBAL_LOAD_ASYNC_TO_LDS_B32) -----
#if defined(__AMDGCN__) && \
    __has_builtin(__builtin_amdgcn_global_load_async_to_lds_b32) && \
    __has_builtin(__builtin_amdgcn_s_wait_asynccnt)
#define HAS_ASYNC_LDS 1
typedef __attribute__((address_space(1))) int async_gint;
typedef __attribute__((address_space(3))) int async_lint;
static __device__ __forceinline__ void async_copy_b32(const float* g, float* l) {
  __builtin_amdgcn_global_load_async_to_lds_b32(
      (async_gint*)g, (async_lint*)l, /*imm offset=*/0, /*cpol=*/0);
}
static __device__ __forceinline__ void async_wait0() {
  __builtin_amdgcn_s_wait_asynccnt(0);
}
#else
#define HAS_ASYNC_LDS 0
static __device__ __forceinline__ void async_copy_b32(const float* g, float* l) {
  *l = *g;
}
static __device__ __forceinline__ void async_wait0() {}
#endif

// ---------------- utility ----------------
__global__ void zero_f32_kernel(float* __restrict__ p, int n) {
  int i = blockIdx.x * blockDim.x + threadIdx.x;
  if (i < n) p[i] = 0.0f;
}

// ---------------- phase 1: LSTM weight evolution ----------------
// One block per output element j (0..FLAT-1); 4 waves compute the 4 gate rows
// r = j + FLAT*gate via a 1024-long dot against inp and h (float4 streaming
// loads -> global_load_b128), shuffle-reduced.
__global__ __launch_bounds__(128)
void lstm_step_kernel(const float* __restrict__ inp,
                      const float* __restrict__ hin,
                      const float* __restrict__ cin,
                      const float* __restrict__ W_ih,
                      const float* __restrict__ W_hh,
                      const float* __restrict__ b_ih,
                      const float* __restrict__ b_hh,
                      float* __restrict__ hout,
                      float* __restrict__ cout) {
  const int j    = blockIdx.x;            // 0..FLAT-1
  const int wave = threadIdx.x >> 5;      // gate index 0..3  (i,f,g,o)
  const int lane = threadIdx.x & 31;
  const int r    = j + FLAT * wave;       // row in the [4*FLAT, FLAT] weights

  const float4* wi4 = (const float4*)(W_ih + (size_t)r * FLAT);
  const float4* wh4 = (const float4*)(W_hh + (size_t)r * FLAT);
  const float4* in4 = (const float4*)inp;
  const float4* h4  = (const float4*)hin;

  float s = 0.0f;
  #pragma unroll
  for (int i = 0; i < FLAT / 128; ++i) {    // 8 iters of 4x b128 per lane
    const int idx = lane + 32 * i;
    float4 a = wi4[idx], b = in4[idx];
    float4 c = wh4[idx], d = h4[idx];
    s = fmaf(a.x, b.x, s); s = fmaf(a.y, b.y, s);
    s = fmaf(a.z, b.z, s); s = fmaf(a.w, b.w, s);
    s = fmaf(c.x, d.x, s); s = fmaf(c.y, d.y, s);
    s = fmaf(c.z, d.z, s); s = fmaf(c.w, d.w, s);
  }
  #pragma unroll
  for (int o = 16; o > 0; o >>= 1) s += __shfl_down(s, o, 32);

  __shared__ float g[4];
  if (lane == 0) g[wave] = s + b_ih[r] + b_hh[r];
  __syncthreads();

  if (threadIdx.x == 0) {
    float ig = sigmoidf_(g[0]);
    float fg = sigmoidf_(g[1]);
    float gg = tanhf(g[2]);
    float og = sigmoidf_(g[3]);
    float c2 = fg * cin[j] + ig * gg;
    cout[j] = c2;
    hout[j] = og * tanhf(c2);
  }
}

// ---------------- phase 2a: degrees (self loops included) ----------------
__global__ void degree_kernel(const long long* __restrict__ ei,
                              float* __restrict__ deg) {
  int e = blockIdx.x * blockDim.x + threadIdx.x;
  const int M = NE + NN;
  if (e >= M) return;
  int c = (e < NE) ? (int)ei[NE + e] : (e - NE);   // targets, then self loops
  atomicAdd(&deg[c], 1.0f);
}

__global__ void dinv_kernel(const float* __restrict__ deg,
                            float* __restrict__ dinv) {
  int n = blockIdx.x * blockDim.x + threadIdx.x;
  if (n >= NN) return;
  float d = deg[n];
  dinv[n] = (d > 0.0f) ? rsqrtf(d) : 0.0f;
}

// ---------------- phase 2b: xw = x[-1] @ W via fp32 WMMA ----------------
// One wave per 16-node tile; N=32 output cols as two 16x16 accumulators;
// K=32 consumed in 8 steps of K=4.
__global__ __launch_bounds__(256)
void xw_wmma_kernel(const float* __restrict__ xlast,   // [NN, DIN]
                    const float* __restrict__ W,       // [DIN, DH] row-major
                    float* __restrict__ xw) {          // [NN, DH]
  const int wave  = threadIdx.x >> 5;
  const int lane  = threadIdx.x & 31;
  const int tile  = blockIdx.x * 8 + wave;             // wave-uniform
  const int NTILE = NN / 16;                           // 6250 exact
  if (tile >= NTILE) return;

  const int rb  = tile * 16;
  const int m   = lane & 15;
  const int hi  = lane >> 4;
  const int ko  = hi * 2;                              // K sub-offset per half
  const float* xr = xlast + (size_t)(rb + m) * DIN;

  v8f acc0 = {};                                       // cols 0..15
  v8f acc1 = {};                                       // cols 16..31
  #pragma unroll
  for (int kk = 0; kk < 8; ++kk) {
    const int kb = 4 * kk + ko;
    v2f a;  a.x = xr[kb];                a.y = xr[kb + 1];
    v2f b0; b0.x = W[kb * DH + m];       b0.y = W[(kb + 1) * DH + m];
    v2f b1; b1.x = W[kb * DH + 16 + m];  b1.y = W[(kb + 1) * DH + 16 + m];
    acc0 = wmma4(a, b0, acc0);
    acc1 = wmma4(a, b1, acc1);
  }

  #pragma unroll
  for (int j = 0; j < 8; ++j) {
    const int row = rb + j + 8 * hi;                   // D layout (§7.12.2)
    xw[(size_t)row * DH + m]       = acc0[j];
    xw[(size_t)row * DH + 16 + m]  = acc1[j];
  }
}

// ---------------- phase 2c: normalized scatter-add ----------------
// thread = (edge, column) pair; 32 consecutive lanes cover one edge row.
__global__ void scatter_kernel(const long long* __restrict__ ei,
                               const float* __restrict__ dinv,
                               const float* __restrict__ xw,
                               float* __restrict__ agg) {
  long long t = (long long)blockIdx.x * blockDim.x + threadIdx.x;
  const long long TOT = (long long)(NE + NN) * 32;
  if (t >= TOT) return;
  int e = (int)(t >> 5);
  int l = (int)(t & 31);
  int r, c;
  if (e < NE) { r = (int)ei[e]; c = (int)ei[NE + e]; }
  else        { r = e - NE;     c = e - NE; }
  float nv = dinv[r] * dinv[c];
  atomicAdd(&agg[(size_t)c * DH + l], nv * xw[(size_t)r * DH + l]);
}

// ---------------- phase 3: edge MLP via fp32 WMMA ----------------
// One wave per 16-edge tile. edge_in tile (K=80 x N=16) staged in LDS in
// [K][N] order via coalesced, per-row ASYNC-to-LDS copies. W1 (32x80) staged
// block-wide. D = W1 * edge_in^T computed as 2 M-tiles x 20 K-steps of
// wmma_f32_16x16x4_f32. ReLU of the GCN output is folded into the B-fragment
// reads (K-steps 0..15 = emb rows; 16..19 = edge_attr rows, no ReLU).
// Epilogue: ReLU+b1, dot with W2 per-lane, shfl_xor(16) combine, +b2.
__global__ __launch_bounds__(256)
void edge_mlp_kernel(const long long* __restrict__ ei,
                     const float* __restrict__ agg,     // [NN, DH], pre-ReLU
                     const float* __restrict__ eattr,   // [NE, EF]
                     const float* __restrict__ W1,      // [DH, KIN] row-major
                     const float* __restrict__ b1,
                     const float* __restrict__ W2,      // [1, DH]
                     const float* __restrict__ b2,
                     float* __restrict__ out) {         // [NE]
  __shared__ float sW1[DH * KIN];          // 2560
  __shared__ float sB[8 * KIN * 16];       // per-wave 80x16 edge_in tiles
  __shared__ float sB1[DH];
  __shared__ float sW2[DH];

  const int wave = threadIdx.x >> 5;
  const int lane = threadIdx.x & 31;
  const int tile = blockIdx.x * 8 + wave;              // wave-uniform
  const int NTILE = NE / 16;                           // 100000 exact

  // block-wide W1 staging (coalesced b32, async where available)
  #pragma unroll
  for (int i = 0; i < (DH * KIN) / 256; ++i)
    async_copy_b32(&W1[threadIdx.x + 256 * i], &sW1[threadIdx.x + 256 * i]);
  if (threadIdx.x < DH)      sB1[threadIdx.x] = b1[threadIdx.x];
  else if (threadIdx.x < 64) sW2[threadIdx.x - 32] = W2[threadIdx.x - 32];

  float* sBw = sB + wave * (KIN * 16);
  const int n  = lane & 15;
  const int hi = lane >> 4;

  if (tile < NTILE) {
    const int ebase = tile * 16;
    // emb[src_i] rows: lane = feature k, one coalesced 128B row per edge
    #pragma unroll
    for (int i = 0; i < 16; ++i) {
      const float* srow = agg + (size_t)((int)ei[ebase + i]) * DH;
      async_copy_b32(&srow[lane], &sBw[lane * 16 + i]);
    }
    // emb[dst_i] rows
    #pragma unroll
    for (int i = 0; i < 16; ++i) {
      const float* drow = agg + (size_t)((int)ei[NE + ebase + i]) * DH;
      async_copy_b32(&drow[lane], &sBw[(DH + lane) * 16 + i]);
    }
    // edge_attr rows: 2 edges per iteration (16 lanes each), full EXEC
    #pragma unroll
    for (int i = 0; i < 8; ++i) {
      const int e2 = ebase + 2 * i + hi;
      async_copy_b32(&eattr[(size_t)e2 * EF + n],
                     &sBw[(2 * DH + n) * 16 + (2 * i + hi)]);
    }
  }
  async_wait0();
  __syncthreads();

  if (tile >= NTILE) return;

  const int ko = hi * 2;
  v8f acc0 = {};                                       // rows m = 0..15
  v8f acc1 = {};                                       // rows m = 16..31
  #pragma unroll
  for (int kk = 0; kk < 20; ++kk) {
    const int kb = 4 * kk + ko;
    v2f a0; a0.x = sW1[n * KIN + kb];         a0.y = sW1[n * KIN + kb + 1];
    v2f a1; a1.x = sW1[(16 + n) * KIN + kb];  a1.y = sW1[(16 + n) * KIN + kb + 1];
    v2f b;  b.x  = sBw[kb * 16 + n];          b.y  = sBw[(kb + 1) * 16 + n];
    if (kk < 16) {                 // emb rows: apply the GCN ReLU here
      b.x = fmaxf(b.x, 0.0f);
      b.y = fmaxf(b.y, 0.0f);
    }
    acc0 = wmma4(a0, b, acc0);
    acc1 = wmma4(a1, b, acc1);
  }

  // epilogue: this lane holds column n, rows {8hi..8hi+7} U {16+8hi..16+8hi+7}
  const float b2v = b2[0];
  float partial = 0.0f;
  #pragma unroll
  for (int j = 0; j < 8; ++j) {
    const int m0 = j + 8 * hi;
    const int m1 = 16 + j + 8 * hi;
    partial = fmaf(sW2[m0], fmaxf(acc0[j] + sB1[m0], 0.0f), partial);
    partial = fmaf(sW2[m1], fmaxf(acc1[j] + sB1[m1], 0.0f), partial);
  }
  partial += __shfl_xor(partial, 16, 32);
  if (hi == 0) out[tile * 16 + n] = partial + b2v;
}

// ---------------- host-side orchestration ----------------
extern "C" void kernel_launch(void* const* d_in, const int* in_sizes, int n_in,
                              void* d_out, int out_size, void* d_ws, size_t ws_size,
                              hipStream_t stream) {
  const float*     x      = (const float*)d_in[0];
  const long long* ei     = (const long long*)d_in[1];   // int64 [2, NE]
  const float*     eattr  = (const float*)d_in[2];
  const float*     W_ih   = (const float*)d_in[3];
  const float*     W_hh   = (const float*)d_in[4];
  const float*     b_ih   = (const float*)d_in[5];
  const float*     b_hh   = (const float*)d_in[6];
  const float*     init_w = (const float*)d_in[7];
  const float*     W1     = (const float*)d_in[8];
  const float*     b1     = (const float*)d_in[9];
  const float*     W2     = (const float*)d_in[10];
  const float*     b2     = (const float*)d_in[11];
  float*           out    = (float*)d_out;

  // workspace carve (floats)
  float* ws = (float*)d_ws;
  size_t o = 0;
  float* zbuf = ws + o; o += FLAT;        // zeros for h0/c0
  float* hb0  = ws + o; o += FLAT;
  float* hb1  = ws + o; o += FLAT;
  float* cb0  = ws + o; o += FLAT;
  float* cb1  = ws + o; o += FLAT;
  float* deg  = ws + o; o += NN;
  float* dinv = ws + o; o += NN;
  float* xw   = ws + o; o += (size_t)NN * DH;
  float* agg  = ws + o; o += (size_t)NN * DH;

  // zero-init buffers that are read before being fully written
  zero_f32_kernel<<<(FLAT + 255) / 256, 256, 0, stream>>>(zbuf, FLAT);
  zero_f32_kernel<<<(NN + 255) / 256, 256, 0, stream>>>(deg, NN);
  zero_f32_kernel<<<((NN * DH) + 255) / 256, 256, 0, stream>>>(agg, NN * DH);

  // --- LSTM: 5 sequential steps, ping-pong h/c (inp == h after step 0) ---
  lstm_step_kernel<<<FLAT, 128, 0, stream>>>(init_w, zbuf, zbuf,
                                             W_ih, W_hh, b_ih, b_hh, hb0, cb0);
  lstm_step_kernel<<<FLAT, 128, 0, stream>>>(hb0, hb0, cb0,
                                             W_ih, W_hh, b_ih, b_hh, hb1, cb1);
  lstm_step_kernel<<<FLAT, 128, 0, stream>>>(hb1, hb1, cb1,
                                             W_ih, W_hh, b_ih, b_hh, hb0, cb0);
  lstm_step_kernel<<<FLAT, 128, 0, stream>>>(hb0, hb0, cb0,
                                             W_ih, W_hh, b_ih, b_hh, hb1, cb1);
  lstm_step_kernel<<<FLAT, 128, 0, stream>>>(hb1, hb1, cb1,
                                             W_ih, W_hh, b_ih, b_hh, hb0, cb0);
  const float* Wev = hb0;  // evolved GNN weight, [DIN, DH] flattened

  // --- GCN normalization ---
  const int M = NE + NN;
  degree_kernel<<<(M + 255) / 256, 256, 0, stream>>>(ei, deg);
  dinv_kernel<<<(NN + 255) / 256, 256, 0, stream>>>(deg, dinv);

  // --- xw = x[-1] @ W (WMMA) ---
  const float* xlast = x + (size_t)(TT - 1) * NN * DIN;
  xw_wmma_kernel<<<(NN / 16 + 7) / 8, 256, 0, stream>>>(xlast, Wev, xw);

  // --- message passing: agg[c] += norm * xw[r]  (ReLU folded into edge MLP) ---
  long long tot = (long long)M * 32;
  scatter_kernel<<<(int)((tot + 255) / 256), 256, 0, stream>>>(ei, dinv, xw, agg);

  // --- edge MLP (WMMA + async LDS staging) ---
  edge_mlp_kernel<<<NE / 16 / 8, 256, 0, stream>>>(ei, agg, eattr,
                                                   W1, b1, W2, b2, out);
  (void)in_sizes; (void)n_in; (void)out_size; (void)ws_size;
}